// ANEGCN_2_82721070121489
// MI455X (gfx1250) — compile-verified
//
#include <hip/hip_runtime.h>
#include <hip/hip_bf16.h>

#define RDIM 116
#define RR   (RDIM * RDIM)
#define BATCH 1024
#define NLAYER 4
#define EPSBN 1e-5f

typedef __attribute__((ext_vector_type(2))) float f32x2;
typedef __attribute__((ext_vector_type(8))) float f32x8;

// ---------------------------------------------------------------------------
// Batched 116x116x116 fp32 GEMM on WMMA F32 16x16x4. One block per batch
// element; 8 waves; wave w computes rows [16w,16w+16) x all 128 padded cols.
// Explicit two-stage double buffer, manually unrolled x2: each fragment
// buffer is written directly by global loads (no shift-register movs), so
// the scheduler emits load-clause -> partial wait -> 8 WMMAs per stage.
// TA/TB: operand transposed (index-swapped load). APB/BPB: per-batch vs
// shared weights. BIAS: 0 none, 1 per-row, 2 per-col.
// ---------------------------------------------------------------------------
template <bool TA, bool TB, bool APB, bool BPB, int BIAS, bool RELU>
__global__ __launch_bounds__(256) void gemm116(const float* __restrict__ A,
                                               const float* __restrict__ Bm,
                                               const float* __restrict__ bias,
                                               float* __restrict__ C) {
  const int b = blockIdx.x;
  const int wave = threadIdx.x >> 5;
  const int lane = threadIdx.x & 31;
  const int lane_lo = lane & 15;
  const int lane_hi = lane >> 4;

  const float* __restrict__ Ap = APB ? A + (size_t)b * RR : A;
  const float* __restrict__ Bp = BPB ? Bm + (size_t)b * RR : Bm;
  float* __restrict__ Cp = C + (size_t)b * RR;

  const int m0 = wave * 16;
  int arow = m0 + lane_lo;
  if (arow > RDIM - 1) arow = RDIM - 1;  // clamp padded rows (masked at store)

  int bcolv[8];
#pragma unroll
  for (int t = 0; t < 8; ++t) {
    int c = t * 16 + lane_lo;
    bcolv[t] = (c > RDIM - 1) ? (RDIM - 1) : c;
  }

  auto loadFrags = [&](int k0, f32x2& af, f32x2 (&bf)[8]) {
    const int kk = k0 + 2 * lane_hi;
    if (TA) {
      af.x = Ap[kk * RDIM + arow];
      af.y = Ap[(kk + 1) * RDIM + arow];
    } else {
      af.x = Ap[arow * RDIM + kk];
      af.y = Ap[arow * RDIM + kk + 1];
    }
#pragma unroll
    for (int t = 0; t < 8; ++t) {
      if (TB) {
        bf[t].x = Bp[bcolv[t] * RDIM + kk];
        bf[t].y = Bp[bcolv[t] * RDIM + kk + 1];
      } else {
        bf[t].x = Bp[kk * RDIM + bcolv[t]];
        bf[t].y = Bp[(kk + 1) * RDIM + bcolv[t]];
      }
    }
  };

  f32x8 acc[8];
#pragma unroll
  for (int t = 0; t < 8; ++t) acc[t] = f32x8{};

  auto doWmma = [&](const f32x2& af, const f32x2 (&bf)[8]) {
#pragma unroll
    for (int t = 0; t < 8; ++t)
      acc[t] = __builtin_amdgcn_wmma_f32_16x16x4_f32(false, af, false, bf[t],
                                                     (short)0, acc[t], false,
                                                     false);
  };

  // 29 k-steps of 4 (116 = 29*4). 14 double iterations + 1 tail compute.
  f32x2 af0, bf0[8], af1, bf1[8];
  loadFrags(0, af0, bf0);
  int k0 = 0;
  for (int it = 0; it < 14; ++it) {
    loadFrags(k0 + 4, af1, bf1);
    doWmma(af0, bf0);
    loadFrags(k0 + 8, af0, bf0);  // last prefetch: k=112 (in range)
    doWmma(af1, bf1);
    k0 += 8;
  }
  doWmma(af0, bf0);  // k = 112

#pragma unroll
  for (int t = 0; t < 8; ++t) {
    const int col = t * 16 + lane_lo;
    if (col >= RDIM) continue;
    const float cbias = (BIAS == 2) ? bias[col] : 0.f;
#pragma unroll
    for (int v = 0; v < 8; ++v) {
      const int row = m0 + v + 8 * lane_hi;
      if (row >= RDIM) continue;
      float val = acc[t][v] + cbias + ((BIAS == 1) ? bias[row] : 0.f);
      if (RELU) val = fmaxf(val, 0.f);
      Cp[row * RDIM + col] = val;
    }
  }
}

// ---------------------------------------------------------------------------
// Classifier GEMM: C[M,N] = relu(A[M,K] @ B[N,K]^T + bias[N]).
// M=N=1024, K=2320 (all divisible: 128 / 8). grid=(M/128, N/128), 8 waves.
// Same double-buffered structure; K compile-time so trip counts are exact.
// ---------------------------------------------------------------------------
template <int M, int N, int K>
__global__ __launch_bounds__(256) void gemm_cls(const float* __restrict__ A,
                                                const float* __restrict__ Bw,
                                                const float* __restrict__ bias,
                                                float* __restrict__ C) {
  const int wave = threadIdx.x >> 5;
  const int lane = threadIdx.x & 31;
  const int lane_lo = lane & 15;
  const int lane_hi = lane >> 4;
  const int m0 = blockIdx.x * 128 + wave * 16;
  const int n0 = blockIdx.y * 128;

  const float* __restrict__ Arow = A + (size_t)(m0 + lane_lo) * K;
  const float* __restrict__ Brow[8];
#pragma unroll
  for (int t = 0; t < 8; ++t)
    Brow[t] = Bw + (size_t)(n0 + t * 16 + lane_lo) * K;

  auto loadFrags = [&](int k0, f32x2& af, f32x2 (&bf)[8]) {
    const int kk = k0 + 2 * lane_hi;
    af.x = Arow[kk];
    af.y = Arow[kk + 1];
#pragma unroll
    for (int t = 0; t < 8; ++t) {
      bf[t].x = Brow[t][kk];
      bf[t].y = Brow[t][kk + 1];
    }
  };

  f32x8 acc[8];
#pragma unroll
  for (int t = 0; t < 8; ++t) acc[t] = f32x8{};

  auto doWmma = [&](const f32x2& af, const f32x2 (&bf)[8]) {
#pragma unroll
    for (int t = 0; t < 8; ++t)
      acc[t] = __builtin_amdgcn_wmma_f32_16x16x4_f32(false, af, false, bf[t],
                                                     (short)0, acc[t], false,
                                                     false);
  };

  // K/4 k-steps; (K/8 - 1) double iterations, then tail: k = K-8 and K-4.
  f32x2 af0, bf0[8], af1, bf1[8];
  loadFrags(0, af0, bf0);
  int k0 = 0;
  for (int it = 0; it < K / 8 - 1; ++it) {
    loadFrags(k0 + 4, af1, bf1);
    doWmma(af0, bf0);
    loadFrags(k0 + 8, af0, bf0);
    doWmma(af1, bf1);
    k0 += 8;
  }
  loadFrags(K - 4, af1, bf1);
  doWmma(af0, bf0);  // k = K-8
  doWmma(af1, bf1);  // k = K-4

#pragma unroll
  for (int t = 0; t < 8; ++t) {
    const int col = n0 + t * 16 + lane_lo;
    const float cb = bias[col];
#pragma unroll
    for (int v = 0; v < 8; ++v) {
      const int row = m0 + v + 8 * lane_hi;
      C[(size_t)row * N + col] = fmaxf(acc[t][v] + cb, 0.f);
    }
  }
}

// ---------------------------------------------------------------------------
// Attention: K = aw1 @ X^T + ab1 (3xR), scores = K^T K, softmax rows -> att.
// One block per batch, thread n owns row n. K staged in LDS, 3-pass softmax.
// ---------------------------------------------------------------------------
__global__ __launch_bounds__(128) void attn_kernel(const float* __restrict__ X,
                                                   const float* __restrict__ aw1,
                                                   const float* __restrict__ ab1,
                                                   float* __restrict__ att) {
  __shared__ float Ks[3][RDIM];
  const int b = blockIdx.x;
  const int n = threadIdx.x;
  if (n < RDIM) {
    const float x0 = X[(b * RDIM + n) * 3 + 0];
    const float x1 = X[(b * RDIM + n) * 3 + 1];
    const float x2 = X[(b * RDIM + n) * 3 + 2];
#pragma unroll
    for (int c = 0; c < 3; ++c)
      Ks[c][n] = aw1[c * 3 + 0] * x0 + aw1[c * 3 + 1] * x1 +
                 aw1[c * 3 + 2] * x2 + ab1[c];
  }
  __syncthreads();
  if (n >= RDIM) return;
  const float k0 = Ks[0][n], k1 = Ks[1][n], k2 = Ks[2][n];
  float mx = -3.0e38f;
  for (int m = 0; m < RDIM; ++m) {
    const float s = k0 * Ks[0][m] + k1 * Ks[1][m] + k2 * Ks[2][m];
    mx = fmaxf(mx, s);
  }
  float sum = 0.f;
  for (int m = 0; m < RDIM; ++m) {
    const float s = k0 * Ks[0][m] + k1 * Ks[1][m] + k2 * Ks[2][m];
    sum += __expf(s - mx);
  }
  const float inv = 1.f / sum;
  float* __restrict__ row = att + ((size_t)b * RDIM + n) * RDIM;
  for (int m = 0; m < RDIM; ++m) {
    const float s = k0 * Ks[0][m] + k1 * Ks[1][m] + k2 * Ks[2][m];
    row[m] = __expf(s - mx) * inv;
  }
}

// ---------------------------------------------------------------------------
// BatchNorm stats. Edge: channel n of Z1(B,n,m), stats over (B,m).
// Node: channel n of X1(B,n,c), stats over (B,3). mv[2n]=mean, mv[2n+1]=rstd.
// ---------------------------------------------------------------------------
__global__ __launch_bounds__(256) void bn_edge_stats(const float* __restrict__ Z1,
                                                     float* __restrict__ mv) {
  const int n = blockIdx.x;
  float s = 0.f, q = 0.f;
  for (int idx = threadIdx.x; idx < BATCH * RDIM; idx += 256) {
    const int bb = idx / RDIM, m = idx % RDIM;
    const float v = Z1[(size_t)bb * RR + n * RDIM + m];
    s += v;
    q += v * v;
  }
  __shared__ float ss[256], qq[256];
  ss[threadIdx.x] = s;
  qq[threadIdx.x] = q;
  __syncthreads();
  for (int o = 128; o > 0; o >>= 1) {
    if (threadIdx.x < o) {
      ss[threadIdx.x] += ss[threadIdx.x + o];
      qq[threadIdx.x] += qq[threadIdx.x + o];
    }
    __syncthreads();
  }
  if (threadIdx.x == 0) {
    const float inv = 1.f / (float)(BATCH * RDIM);
    const float mean = ss[0] * inv;
    const float var = qq[0] * inv - mean * mean;
    mv[2 * n] = mean;
    mv[2 * n + 1] = rsqrtf(var + EPSBN);
  }
}

__global__ __launch_bounds__(256) void bn_node_stats(const float* __restrict__ X1,
                                                     float* __restrict__ mv) {
  const int n = blockIdx.x;
  float s = 0.f, q = 0.f;
  for (int idx = threadIdx.x; idx < BATCH * 3; idx += 256) {
    const int bb = idx / 3, c = idx % 3;
    const float v = X1[((size_t)bb * RDIM + n) * 3 + c];
    s += v;
    q += v * v;
  }
  __shared__ float ss[256], qq[256];
  ss[threadIdx.x] = s;
  qq[threadIdx.x] = q;
  __syncthreads();
  for (int o = 128; o > 0; o >>= 1) {
    if (threadIdx.x < o) {
      ss[threadIdx.x] += ss[threadIdx.x + o];
      qq[threadIdx.x] += qq[threadIdx.x + o];
    }
    __syncthreads();
  }
  if (threadIdx.x == 0) {
    const float inv = 1.f / (float)(BATCH * 3);
    const float mean = ss[0] * inv;
    const float var = qq[0] * inv - mean * mean;
    mv[2 * n] = mean;
    mv[2 * n + 1] = rsqrtf(var + EPSBN);
  }
}

// Zcur = relu(bn(Z1)) + Zcur ; channel = (i/R)%R
__global__ void bn_edge_apply(const float* __restrict__ Z1,
                              const float* __restrict__ mv,
                              const float* __restrict__ g,
                              const float* __restrict__ be,
                              float* __restrict__ Zcur) {
  const size_t total = (size_t)BATCH * RR;
  for (size_t i = (size_t)blockIdx.x * blockDim.x + threadIdx.x; i < total;
       i += (size_t)gridDim.x * blockDim.x) {
    const int n = (int)((i / RDIM) % RDIM);
    const float h = (Z1[i] - mv[2 * n]) * mv[2 * n + 1] * g[n] + be[n];
    Zcur[i] += fmaxf(h, 0.f);
  }
}

// Xcur = relu(bn(X1)) + Xcur ; channel = (i/3)%R
__global__ void bn_node_apply(const float* __restrict__ X1,
                              const float* __restrict__ mv,
                              const float* __restrict__ g,
                              const float* __restrict__ be,
                              float* __restrict__ Xcur) {
  const size_t total = (size_t)BATCH * RDIM * 3;
  for (size_t i = (size_t)blockIdx.x * blockDim.x + threadIdx.x; i < total;
       i += (size_t)gridDim.x * blockDim.x) {
    const int n = (int)((i / 3) % RDIM);
    const float h = (X1[i] - mv[2 * n]) * mv[2 * n + 1] * g[n] + be[n];
    Xcur[i] += fmaxf(h, 0.f);
  }
}

// ---------------------------------------------------------------------------
// Node update: X1[b,n,c] = nw @ (sum_m P[b,m,n] * X[b,m,:]) + nb.
// (reference A[n,m] == P[m,n], so this is A @ X without materializing A^T)
// ---------------------------------------------------------------------------
__global__ __launch_bounds__(128) void node_update(const float* __restrict__ P,
                                                   const float* __restrict__ Xin,
                                                   const float* __restrict__ nw,
                                                   const float* __restrict__ nb,
                                                   float* __restrict__ X1) {
  __shared__ float Xs[RDIM][3];
  const int b = blockIdx.x;
  if (threadIdx.x < RDIM) {
#pragma unroll
    for (int c = 0; c < 3; ++c)
      Xs[threadIdx.x][c] = Xin[(b * RDIM + threadIdx.x) * 3 + c];
  }
  __syncthreads();
  const int n = threadIdx.x;
  if (n >= RDIM) return;
  float s0 = 0.f, s1 = 0.f, s2 = 0.f;
  const float* __restrict__ pc = P + (size_t)b * RR + n;  // P[m][n], stride R
  for (int m = 0; m < RDIM; ++m) {
    const float p = pc[(size_t)m * RDIM];
    s0 += p * Xs[m][0];
    s1 += p * Xs[m][1];
    s2 += p * Xs[m][2];
  }
#pragma unroll
  for (int c = 0; c < 3; ++c)
    X1[(b * RDIM + n) * 3 + c] =
        nw[c * 3 + 0] * s0 + nw[c * 3 + 1] * s1 + nw[c * 3 + 2] * s2 + nb[c];
}

// ---------------------------------------------------------------------------
// ConvDownSample(node): f = relu(X . w + b), fbuf + atomic partial sums.
// ---------------------------------------------------------------------------
__global__ __launch_bounds__(256) void dn_compute(const float* __restrict__ Xin,
                                                  const float* __restrict__ w,
                                                  const float* __restrict__ bsc,
                                                  float* __restrict__ fbuf,
                                                  float* __restrict__ stats) {
  float s = 0.f, q = 0.f;
  for (int idx = blockIdx.x * 256 + threadIdx.x; idx < BATCH * RDIM;
       idx += gridDim.x * 256) {
    const float f = fmaxf(Xin[idx * 3 + 0] * w[0] + Xin[idx * 3 + 1] * w[1] +
                              Xin[idx * 3 + 2] * w[2] + bsc[0],
                          0.f);
    fbuf[idx] = f;
    s += f;
    q += f * f;
  }
  __shared__ float ss[256], qq[256];
  ss[threadIdx.x] = s;
  qq[threadIdx.x] = q;
  __syncthreads();
  for (int o = 128; o > 0; o >>= 1) {
    if (threadIdx.x < o) {
      ss[threadIdx.x] += ss[threadIdx.x + o];
      qq[threadIdx.x] += qq[threadIdx.x + o];
    }
    __syncthreads();
  }
  if (threadIdx.x == 0) {
    atomicAdd(&stats[0], ss[0]);
    atomicAdd(&stats[1], qq[0]);
  }
}

__global__ void dn_apply(const float* __restrict__ fbuf,
                         const float* __restrict__ stats,
                         const float* __restrict__ g,
                         const float* __restrict__ be, float* __restrict__ XZ,
                         int layer) {
  const float inv = 1.f / (float)(BATCH * RDIM);
  const float mean = stats[0] * inv;
  const float var = stats[1] * inv - mean * mean;
  const float rstd = rsqrtf(var + EPSBN);
  for (int idx = blockIdx.x * blockDim.x + threadIdx.x; idx < BATCH * RDIM;
       idx += gridDim.x * blockDim.x) {
    const int b = idx / RDIM, r = idx % RDIM;
    XZ[(size_t)b * 2320 + layer * RDIM + r] =
        (fbuf[idx] - mean) * rstd * g[0] + be[0];
  }
}

// ---------------------------------------------------------------------------
// ConvDownSample(edge): Conv1d(1,1,78,stride=39,pad=39) -> 3 taps per Z row.
// ---------------------------------------------------------------------------
__global__ __launch_bounds__(256) void de_compute(const float* __restrict__ Zin,
                                                  const float* __restrict__ w,
                                                  const float* __restrict__ bsc,
                                                  float* __restrict__ fbuf,
                                                  float* __restrict__ stats) {
  float s = 0.f, q = 0.f;
  for (int idx = blockIdx.x * 256 + threadIdx.x; idx < BATCH * RDIM;
       idx += gridDim.x * 256) {
    const float* __restrict__ row = Zin + (size_t)idx * RDIM;
    float f0 = 0.f, f1 = 0.f, f2 = 0.f;
    for (int j = 0; j < 39; ++j) f0 += w[39 + j] * row[j];   // window t=0
    for (int j = 0; j < 78; ++j) f1 += w[j] * row[j];        // window t=1
    for (int j = 39; j < RDIM; ++j) f2 += w[j - 39] * row[j];// window t=2
    f0 = fmaxf(f0 + bsc[0], 0.f);
    f1 = fmaxf(f1 + bsc[0], 0.f);
    f2 = fmaxf(f2 + bsc[0], 0.f);
    fbuf[idx * 3 + 0] = f0;
    fbuf[idx * 3 + 1] = f1;
    fbuf[idx * 3 + 2] = f2;
    s += f0 + f1 + f2;
    q += f0 * f0 + f1 * f1 + f2 * f2;
  }
  __shared__ float ss[256], qq[256];
  ss[threadIdx.x] = s;
  qq[threadIdx.x] = q;
  __syncthreads();
  for (int o = 128; o > 0; o >>= 1) {
    if (threadIdx.x < o) {
      ss[threadIdx.x] += ss[threadIdx.x + o];
      qq[threadIdx.x] += qq[threadIdx.x + o];
    }
    __syncthreads();
  }
  if (threadIdx.x == 0) {
    atomicAdd(&stats[0], ss[0]);
    atomicAdd(&stats[1], qq[0]);
  }
}

__global__ void de_apply(const float* __restrict__ fbuf,
                         const float* __restrict__ stats,
                         const float* __restrict__ g,
                         const float* __restrict__ be, float* __restrict__ XZ,
                         int layer) {
  const float inv = 1.f / (float)(BATCH * RDIM * 3);
  const float mean = stats[0] * inv;
  const float var = stats[1] * inv - mean * mean;
  const float rstd = rsqrtf(var + EPSBN);
  const int total = BATCH * RDIM * 3;
  for (int idx = blockIdx.x * blockDim.x + threadIdx.x; idx < total;
       idx += gridDim.x * blockDim.x) {
    const int b = idx / (RDIM * 3), rem = idx % (RDIM * 3);
    XZ[(size_t)b * 2320 + 580 + layer * 348 + rem] =
        (fbuf[idx] - mean) * rstd * g[0] + be[0];
  }
}

// Final projection: out[b,:2] = H[b,:] @ cw2^T + cb2
__global__ __launch_bounds__(256) void out_kernel(const float* __restrict__ H,
                                                  const float* __restrict__ w2,
                                                  const float* __restrict__ b2,
                                                  float* __restrict__ out) {
  const int b = blockIdx.x;
  float s0 = 0.f, s1 = 0.f;
  for (int h = threadIdx.x; h < 1024; h += 256) {
    const float hv = H[(size_t)b * 1024 + h];
    s0 += hv * w2[h];
    s1 += hv * w2[1024 + h];
  }
  __shared__ float ss[256], qq[256];
  ss[threadIdx.x] = s0;
  qq[threadIdx.x] = s1;
  __syncthreads();
  for (int o = 128; o > 0; o >>= 1) {
    if (threadIdx.x < o) {
      ss[threadIdx.x] += ss[threadIdx.x + o];
      qq[threadIdx.x] += qq[threadIdx.x + o];
    }
    __syncthreads();
  }
  if (threadIdx.x == 0) {
    out[b * 2 + 0] = ss[0] + b2[0];
    out[b * 2 + 1] = qq[0] + b2[1];
  }
}

__global__ void zero_kernel(float* __restrict__ p, int n) {
  const int i = blockIdx.x * blockDim.x + threadIdx.x;
  if (i < n) p[i] = 0.f;
}

// ---------------------------------------------------------------------------
extern "C" void kernel_launch(void* const* d_in, const int* in_sizes, int n_in,
                              void* d_out, int out_size, void* d_ws,
                              size_t ws_size, hipStream_t stream) {
  (void)in_sizes; (void)n_in; (void)out_size; (void)ws_size;
  const float* X    = (const float*)d_in[0];
  const float* Z    = (const float*)d_in[1];
  const float* aw1  = (const float*)d_in[2];
  const float* ab1  = (const float*)d_in[3];
  const float* aw2  = (const float*)d_in[4];
  const float* ab2  = (const float*)d_in[5];
  const float* nw   = (const float*)d_in[6];
  const float* nb   = (const float*)d_in[7];
  const float* ew   = (const float*)d_in[8];
  const float* eb   = (const float*)d_in[9];
  const float* gn_g = (const float*)d_in[10];
  const float* gn_b = (const float*)d_in[11];
  const float* ge_g = (const float*)d_in[12];
  const float* ge_b = (const float*)d_in[13];
  const float* dn_w = (const float*)d_in[14];
  const float* dn_b = (const float*)d_in[15];
  const float* dn_g = (const float*)d_in[16];
  const float* dn_be= (const float*)d_in[17];
  const float* de_w = (const float*)d_in[18];
  const float* de_b = (const float*)d_in[19];
  const float* de_g = (const float*)d_in[20];
  const float* de_be= (const float*)d_in[21];
  const float* cw1  = (const float*)d_in[22];
  const float* cb1  = (const float*)d_in[23];
  const float* cw2  = (const float*)d_in[24];
  const float* cb2  = (const float*)d_in[25];
  float* out = (float*)d_out;

  // workspace layout (floats)
  float* ws = (float*)d_ws;
  size_t off = 0;
  float* Zcur = ws + off; off += (size_t)BATCH * RR;        // 13.78M
  float* bufA = ws + off; off += (size_t)BATCH * RR;        // att / T
  float* bufB = ws + off; off += (size_t)BATCH * RR;        // V / Z1
  float* bufC = ws + off; off += (size_t)BATCH * RR;        // P
  float* Xcur = ws + off; off += (size_t)BATCH * RDIM * 3;
  float* X1   = ws + off; off += (size_t)BATCH * RDIM * 3;
  float* XZ   = ws + off; off += (size_t)BATCH * 2320;
  float* H    = ws + off; off += (size_t)BATCH * 1024;
  float* fbuf = ws + off; off += (size_t)BATCH * RDIM * 3;
  float* mv_e = ws + off; off += 2 * RDIM;
  float* mv_n = ws + off; off += 2 * RDIM;
  float* stats= ws + off; off += 64;  // dn layer i -> 2i ; de layer i -> 16+2i

  hipMemcpyAsync(Zcur, Z, (size_t)BATCH * RR * sizeof(float),
                 hipMemcpyDeviceToDevice, stream);
  hipMemcpyAsync(Xcur, X, (size_t)BATCH * RDIM * 3 * sizeof(float),
                 hipMemcpyDeviceToDevice, stream);
  zero_kernel<<<1, 64, 0, stream>>>(stats, 64);

  // layer-0 downsamples
  dn_compute<<<464, 256, 0, stream>>>(Xcur, dn_w, dn_b, fbuf, stats + 0);
  dn_apply<<<464, 256, 0, stream>>>(fbuf, stats + 0, dn_g, dn_be, XZ, 0);
  de_compute<<<464, 256, 0, stream>>>(Zcur, de_w, de_b, fbuf, stats + 16);
  de_apply<<<1392, 256, 0, stream>>>(fbuf, stats + 16, de_g, de_be, XZ, 0);

  for (int i = 0; i < NLAYER; ++i) {
    // attention -> bufA
    attn_kernel<<<BATCH, 128, 0, stream>>>(Xcur, aw1 + i * 9, ab1 + i * 3,
                                           bufA);
    // V = aw2 @ Zcur^T + ab2(row)            -> bufB
    gemm116<false, true, false, true, 1, false>
        <<<BATCH, 256, 0, stream>>>(aw2 + (size_t)i * RR, Zcur,
                                    ab2 + i * RDIM, bufB);
    // P = att @ V                             -> bufC
    gemm116<false, false, true, true, 0, false>
        <<<BATCH, 256, 0, stream>>>(bufA, bufB, nullptr, bufC);
    // T = P^T @ Zcur                          -> bufA
    gemm116<true, false, true, true, 0, false>
        <<<BATCH, 256, 0, stream>>>(bufC, Zcur, nullptr, bufA);
    // Z1 = T @ ew^T + eb(col)                 -> bufB
    gemm116<false, true, true, false, 2, false>
        <<<BATCH, 256, 0, stream>>>(bufA, ew + (size_t)i * RR, eb + i * RDIM,
                                    bufB);
    // edge BN + relu + residual into Zcur
    bn_edge_stats<<<RDIM, 256, 0, stream>>>(bufB, mv_e);
    bn_edge_apply<<<4096, 256, 0, stream>>>(bufB, mv_e, ge_g + i * RDIM,
                                            ge_b + i * RDIM, Zcur);
    de_compute<<<464, 256, 0, stream>>>(Zcur, de_w + (i + 1) * 78,
                                        de_b + (i + 1), fbuf,
                                        stats + 16 + 2 * (i + 1));
    de_apply<<<1392, 256, 0, stream>>>(fbuf, stats + 16 + 2 * (i + 1),
                                       de_g + (i + 1), de_be + (i + 1), XZ,
                                       i + 1);
    // node update (uses P and pre-update Xcur)
    node_update<<<BATCH, 128, 0, stream>>>(bufC, Xcur, nw + i * 9, nb + i * 3,
                                           X1);
    bn_node_stats<<<RDIM, 256, 0, stream>>>(X1, mv_n);
    bn_node_apply<<<696, 256, 0, stream>>>(X1, mv_n, gn_g + i * RDIM,
                                           gn_b + i * RDIM, Xcur);
    dn_compute<<<464, 256, 0, stream>>>(Xcur, dn_w + (i + 1) * 3,
                                        dn_b + (i + 1), fbuf,
                                        stats + 2 * (i + 1));
    dn_apply<<<464, 256, 0, stream>>>(fbuf, stats + 2 * (i + 1),
                                      dn_g + (i + 1), dn_be + (i + 1), XZ,
                                      i + 1);
  }

  // classifier: H = relu(XZ @ cw1^T + cb1) ; out = H @ cw2^T + cb2
  gemm_cls<1024, 1024, 2320><<<dim3(8, 8), 256, 0, stream>>>(XZ, cw1, cb1, H);
  out_kernel<<<BATCH, 256, 0, stream>>>(H, cw2, cb2, out);
}